// LearingUpsampler_62251255989036
// MI455X (gfx1250) — compile-verified
//
#include <hip/hip_runtime.h>
#include <hip/hip_bf16.h>
#include <cstdint>

// Learned 2x width upsampler: out[2i] = x[i], out[2i+1] = w*x[i] + (1-w)*x[i+1],
// w = sigmoid(weights0) per feature. Pure streaming op: 128 MiB in + 256 MiB out
// => ~17 us floor at 23.3 TB/s. WMMA is inapplicable (diag matmul would waste
// 15/16 of each tile); the CDNA5 win is the data-movement path:
//   - ONE Tensor Data Mover DMA (tensor_load_to_lds, TENSORcnt) stages the
//     contiguous 33-row tile into LDS; all 8 waves stay free of load issue
//   - conflict-free LDS reads for the blend (64 consecutive lanes read
//     consecutive float4s of one row)
//   - non-temporal b128 stores for the write-once output (> 192 MB L2)

typedef float    v4f  __attribute__((ext_vector_type(4)));
typedef uint32_t u32x4 __attribute__((ext_vector_type(4)));
typedef uint32_t u32x8 __attribute__((ext_vector_type(8)));

constexpr int BATCH   = 16;
constexpr int W       = 8192;
constexpr int F       = 256;        // floats per row
constexpr int OW      = 2 * W - 1;  // 16383
constexpr int TILE    = 32;         // input rows per workgroup
constexpr int VPR     = F / 4;      // float4 per row = 64
constexpr int THREADS = 256;        // 8 wave32

__global__ __launch_bounds__(THREADS)
void LearningUpsampler_kernel(const float* __restrict__ x,
                              const float* __restrict__ w0,
                              float* __restrict__ out)
{
    __shared__ v4f tile[(TILE + 1) * VPR];   // 33 KiB

    const int tid             = threadIdx.x;
    const int tiles_per_batch = W / TILE;                 // 256
    const int bi              = blockIdx.x / tiles_per_batch;
    const int t               = blockIdx.x % tiles_per_batch;
    const int i0              = t * TILE;
    const bool has_halo       = (i0 + TILE) < W;          // last tile: no halo row
    const uint32_t nfl        = (uint32_t)(TILE + (has_halo ? 1 : 0)) * F; // floats in tile

    // Tile rows are contiguous in memory: one flat span of nfl floats.
    const float* src = x + (size_t)(bi * W + i0) * F;

    // Low 32 bits of a flat LDS pointer = byte offset in the workgroup's LDS
    // allocation (LDS aperture lives in addr[63:32]).
    const uint32_t lds_base = (uint32_t)(uintptr_t)(&tile[0]);

    // ---- TDM: one DMA for the whole tile, issued by wave 0 only (TDM ignores
    // EXEC, so every executing wave would launch its own duplicate DMA). ----
    if (tid < 32) {
        const uint64_t ga = (uint64_t)(uintptr_t)src;

        // D# group 0 (128b): count=1 | lds_addr | global_addr[56:0] | type=2
        u32x4 g0;
        g0.x = 0x1u;                                   // count=1, user descriptor
        g0.y = lds_base;                               // LDS byte address
        g0.z = (uint32_t)ga;                           // global_addr[31:0]
        g0.w = ((uint32_t)(ga >> 32) & 0x01FFFFFFu)    // global_addr[56:32]
             | (2u << 30);                             // type = 2 ("image")

        // D# group 1 (256b): 1-row tile of nfl 4-byte elements.
        u32x8 g1;
        g1.s0 = (2u << 16);                            // data_size=4B; wg_mask=0 (no cluster)
        g1.s1 = (nfl & 0xFFFFu) << 16;                 // tensor_dim0[15:0]  @ bits 63:48
        g1.s2 = ((nfl >> 16) & 0xFFFFu)                // tensor_dim0[31:16] @ bits 79:64
              | (1u << 16);                            // tensor_dim1 = 1    @ bits 111:80
        g1.s3 = (nfl & 0xFFFFu) << 16;                 // tile_dim0 = nfl    @ bits 127:112 (8448 max, fits)
        g1.s4 = 1u;                                    // tile_dim1 = 1; tile_dim2 = 0
        g1.s5 = nfl;                                   // tensor_dim0_stride[31:0]
        g1.s6 = 0u;                                    // stride0[47:32]=0; stride1[15:0]=0
        g1.s7 = 0u;                                    // tensor_dim1_stride rest

        asm volatile("tensor_load_to_lds %0, %1"
                     :: "s"(g0), "s"(g1)
                     : "memory");
        __builtin_amdgcn_s_wait_tensorcnt(0);          // DMA done -> LDS visible
    }
    __syncthreads();                                   // release the other 7 waves

    // Per-thread gate for its 4 features (computed once; negligible cost).
    const int v = tid & (VPR - 1);
    v4f wv = ((const v4f*)w0)[v];
    v4f wa;
    wa.x = 1.0f / (1.0f + __expf(-wv.x));
    wa.y = 1.0f / (1.0f + __expf(-wv.y));
    wa.z = 1.0f / (1.0f + __expf(-wv.z));
    wa.w = 1.0f / (1.0f + __expf(-wv.w));
    v4f wb = 1.0f - wa;

    v4f* __restrict__ obase = (v4f*)out + (size_t)bi * OW * VPR;

    const int r0 = tid >> 6;                 // 0..3: row within 4-row stripe
    #pragma unroll
    for (int it = 0; it < TILE / 4; ++it) {
        const int r = it * 4 + r0;           // 0..31
        const int i = i0 + r;
        v4f a = tile[r * VPR + v];

        // Even output row: copy of x. Streaming store (write-once, > L2).
        __builtin_nontemporal_store(a, &obase[(size_t)(2 * i) * VPR + v]);

        if (i < W - 1) {                     // odd rows exist for i in [0, W-2]
            v4f b     = tile[(r + 1) * VPR + v];   // halo row covers r == TILE-1
            v4f inter = wa * a + wb * b;
            __builtin_nontemporal_store(inter, &obase[(size_t)(2 * i + 1) * VPR + v]);
        }
    }
}

extern "C" void kernel_launch(void* const* d_in, const int* in_sizes, int n_in,
                              void* d_out, int out_size, void* d_ws, size_t ws_size,
                              hipStream_t stream)
{
    const float* x  = (const float*)d_in[0];   // [16,1,8192,256] f32
    const float* w0 = (const float*)d_in[1];   // [256] f32
    float* out      = (float*)d_out;           // [16,1,16383,256] f32

    const int tiles_per_batch = W / TILE;        // 256
    const int nblocks = BATCH * tiles_per_batch; // 4096
    LearningUpsampler_kernel<<<nblocks, THREADS, 0, stream>>>(x, w0, out);
}